// MeanFusion_67997922230620
// MI455X (gfx1250) — compile-verified
//
#include <hip/hip_runtime.h>
#include <stdint.h>

#define NUM_GROUPS 100000
#define CHN 128

typedef float v4f __attribute__((ext_vector_type(4)));

// ---------------------------------------------------------------------------
// Kernel 0: zero the workspace (sums[G*C] + counts[G]), vectorized float4.
// ---------------------------------------------------------------------------
__global__ __launch_bounds__(256) void zero_ws_kernel(v4f* __restrict__ p, int n4) {
    int i = blockIdx.x * 256 + threadIdx.x;
    if (i < n4) {
        v4f z = {0.f, 0.f, 0.f, 0.f};
        p[i] = z;
    }
}

// ---------------------------------------------------------------------------
// Kernel 1: scatter-add. One wave32 per point: lane l owns channels [4l,4l+4).
// feat streamed with NT hint (used once); accumulator stays hot in 192MB L2.
// ---------------------------------------------------------------------------
__global__ __launch_bounds__(256) void scatter_kernel(const float* __restrict__ feat,
                                                      const int* __restrict__ gid,
                                                      float* __restrict__ sums,
                                                      float* __restrict__ counts,
                                                      int n) {
    int point = (blockIdx.x * 256 + threadIdx.x) >> 5;   // one wave per point
    int lane  = threadIdx.x & 31;
    if (point >= n) return;

    int g = gid[point];

    const v4f* f4 = (const v4f*)feat;
    v4f v = __builtin_nontemporal_load(&f4[(size_t)point * (CHN / 4) + lane]);

    float* dst = sums + (size_t)g * CHN + lane * 4;
    __hip_atomic_fetch_add(dst + 0, v.x, __ATOMIC_RELAXED, __HIP_MEMORY_SCOPE_AGENT);
    __hip_atomic_fetch_add(dst + 1, v.y, __ATOMIC_RELAXED, __HIP_MEMORY_SCOPE_AGENT);
    __hip_atomic_fetch_add(dst + 2, v.z, __ATOMIC_RELAXED, __HIP_MEMORY_SCOPE_AGENT);
    __hip_atomic_fetch_add(dst + 3, v.w, __ATOMIC_RELAXED, __HIP_MEMORY_SCOPE_AGENT);

    if (lane == 0) {
        __hip_atomic_fetch_add(&counts[g], 1.0f, __ATOMIC_RELAXED, __HIP_MEMORY_SCOPE_AGENT);
    }
}

// ---------------------------------------------------------------------------
// Kernel 2: in-place normalize: sums[g,:] *= 1/max(counts[g],1).
// One wave32 per group row (32 lanes x float4 = 128 channels). L2-resident.
// ---------------------------------------------------------------------------
__global__ __launch_bounds__(256) void normalize_kernel(float* __restrict__ sums,
                                                        const float* __restrict__ counts,
                                                        int g_total) {
    int g    = (blockIdx.x * 256 + threadIdx.x) >> 5;
    int lane = threadIdx.x & 31;
    if (g >= g_total) return;

    float r = 1.0f / fmaxf(counts[g], 1.0f);
    v4f* row = (v4f*)(sums + (size_t)g * CHN);
    v4f v = row[lane];
    v = v * r;
    row[lane] = v;
}

// ---------------------------------------------------------------------------
// Kernel 3: gather via CDNA5 async DMA. One wave32 per point:
//   mean[g] row (512B, L2 hit) --global_load_async_to_lds_b128--> LDS slice
//   LDS slice --global_store_async_from_lds_b128--> out[i] row.
// Tracked with ASYNCcnt; s_wait_asynccnt 0 between load and store covers the
// LDS ordering hazard (async LDS writes may complete out of order).
// Each wave owns a disjoint 512B LDS slice, so no cross-wave hazards.
// ---------------------------------------------------------------------------
__global__ __launch_bounds__(256) void gather_async_kernel(const float* __restrict__ mean,
                                                           const int* __restrict__ gid,
                                                           float* __restrict__ out,
                                                           int n) {
    __shared__ float smem[256 * 4];   // 4KB: 512B per wave, 8 waves per block

    int wave_in_blk = threadIdx.x >> 5;
    int lane        = threadIdx.x & 31;
    int point       = blockIdx.x * 8 + wave_in_blk;
    if (point >= n) return;

    int g = gid[point];

    // addrspace(3) offset of this lane's 16B chunk within the WG's LDS alloc
    unsigned lds_off = (unsigned)(uintptr_t)smem
                     + (unsigned)(wave_in_blk * 512 + lane * 16);

    uint64_t src = (uint64_t)(uintptr_t)(mean + (size_t)g * CHN) + (uint64_t)(lane * 16);
    uint64_t dst = (uint64_t)(uintptr_t)(out + (size_t)point * CHN) + (uint64_t)(lane * 16);

    asm volatile("global_load_async_to_lds_b128 %0, %1, off"
                 :: "v"(lds_off), "v"(src) : "memory");
    asm volatile("s_wait_asynccnt 0x0" ::: "memory");
    asm volatile("global_store_async_from_lds_b128 %0, %1, off"
                 :: "v"(dst), "v"(lds_off) : "memory");
    asm volatile("s_wait_asynccnt 0x0" ::: "memory");
}

// ---------------------------------------------------------------------------
// Launch: zero ws -> atomic scatter -> normalize -> async gather.
// Inputs (setup_inputs order): [0] ref_bxyz (unused), [1] ref_feat f32 [N,128],
// [2] group_ids int32 [N]. Output: f32 [N,128].
// Workspace: sums = float[G*C] at offset 0, counts = float[G] after (51.6 MB).
// ---------------------------------------------------------------------------
extern "C" void kernel_launch(void* const* d_in, const int* in_sizes, int n_in,
                              void* d_out, int out_size, void* d_ws, size_t ws_size,
                              hipStream_t stream) {
    const float* feat = (const float*)d_in[1];
    const int*   gid  = (const int*)d_in[2];
    float*       out  = (float*)d_out;

    int n = in_sizes[2];   // N points

    float* sums   = (float*)d_ws;
    float* counts = sums + (size_t)NUM_GROUPS * CHN;

    int n4 = (NUM_GROUPS * CHN + NUM_GROUPS) / 4;
    zero_ws_kernel<<<(n4 + 255) / 256, 256, 0, stream>>>((v4f*)d_ws, n4);

    scatter_kernel<<<(n + 7) / 8, 256, 0, stream>>>(feat, gid, sums, counts, n);

    normalize_kernel<<<(NUM_GROUPS + 7) / 8, 256, 0, stream>>>(sums, counts, NUM_GROUPS);

    gather_async_kernel<<<(n + 7) / 8, 256, 0, stream>>>(sums, gid, out, n);
}